// DigitCaps_335007449616
// MI455X (gfx1250) — compile-verified
//
#include <hip/hip_runtime.h>
#include <hip/hip_bf16.h>
#include <stdint.h>

#define AS1 __attribute__((address_space(1)))
#define AS3 __attribute__((address_space(3)))

typedef int v4i __attribute__((vector_size(4 * sizeof(int))));

constexpr int BB      = 16;
constexpr int NN      = 16384;
constexpr int CC      = 10;
constexpr int DD      = 16;
constexpr int CD      = CC * DD;        // 160
constexpr int CHUNKS  = 64;
constexpr int NPB     = NN / CHUNKS;    // 256 rows per block
constexpr int TILE_N  = 32;             // rows per LDS tile
constexpr int TILES   = NPB / TILE_N;   // 8
constexpr int PSTRIDE = 336;            // 160 S1 + 160 S2 + 10 rs, padded
constexpr float EPSF  = 1e-9f;

// ---------------- CDNA5 async global->LDS helpers ----------------
__device__ __forceinline__ void async_ld_b128(const void* g, void* l) {
#if __has_builtin(__builtin_amdgcn_global_load_async_to_lds_b128)
  __builtin_amdgcn_global_load_async_to_lds_b128(
      (AS1 v4i*)(g), (AS3 v4i*)(l), 0, 0);
#else
  unsigned long long ga = (unsigned long long)(uintptr_t)g;
  unsigned           la = (unsigned)(uintptr_t)(AS3 void*)l;
  asm volatile("global_load_async_to_lds_b128 %0, %1, off" :: "v"(la), "v"(ga) : "memory");
#endif
}

template <int K>
__device__ __forceinline__ void wait_asynccnt() {
#if __has_builtin(__builtin_amdgcn_s_wait_asynccnt)
  __builtin_amdgcn_s_wait_asynccnt(K);
#else
  asm volatile("s_wait_asynccnt %0" :: "i"(K) : "memory");
#endif
}

// Issue one 32-row tile (20480 B) as 5 async b128 ops per thread (256 threads).
__device__ __forceinline__ void issue_tile(const float* gvotes, float* lbuf,
                                           int t, int tid) {
  const char* g = (const char*)(gvotes + (size_t)t * TILE_N * CD);
  char*       l = (char*)lbuf;
#pragma unroll
  for (int q = 0; q < 5; ++q) {
    int u = q * 256 + tid;             // 16-byte unit index, 0..1279
    async_ld_b128(g + (size_t)u * 16, l + (size_t)u * 16);
  }
}

// ---------------- streaming pass: one read of votes per EM iteration ----------------
template <bool ITER0>
__global__ __launch_bounds__(256)
void em_pass(const float* __restrict__ votes,
             const float* __restrict__ act_in,
             const float* __restrict__ miu_p,
             const float* __restrict__ i2s_p,   // 0.5 / sigma_sq
             const float* __restrict__ hls_p,   // 0.5 * sum_d log sigma_sq
             const float* __restrict__ actp_p,  // previous act_out
             float* __restrict__ partials) {
  __shared__ __align__(16) float ldsbuf[2][TILE_N * CD];   // 2 x 20 KB
  __shared__ float p_miu[CD], p_i2s[CD], p_hls[CC], p_act[CC];
  __shared__ float wred[8][330];

  const int tid   = threadIdx.x;
  const int wave  = tid >> 5;
  const int lane  = tid & 31;
  const int half  = lane >> 4;   // which of the 2 rows this wave-step
  const int dd    = lane & 15;   // d index
  const int blk   = blockIdx.x;
  const int b     = blk / CHUNKS;
  const int chunk = blk % CHUNKS;
  const int n0    = chunk * NPB;

  if (!ITER0) {
    for (int i = tid; i < CD; i += 256) {
      p_miu[i] = miu_p[b * CD + i];
      p_i2s[i] = i2s_p[b * CD + i];
    }
    if (tid < CC) {
      p_hls[tid] = hls_p[b * CC + tid];
      p_act[tid] = actp_p[b * CC + tid];
    }
  }

  float s1[CC], s2[CC], rs[CC];
#pragma unroll
  for (int c = 0; c < CC; ++c) { s1[c] = 0.f; s2[c] = 0.f; rs[c] = 0.f; }

  const float* gvotes = votes + ((size_t)b * NN + n0) * CD;

  issue_tile(gvotes, ldsbuf[0], 0, tid);            // prefetch tile 0
  __syncthreads();                                  // params visible

  for (int t = 0; t < TILES; ++t) {
    if (t + 1 < TILES) {
      issue_tile(gvotes, ldsbuf[(t + 1) & 1], t + 1, tid);
      wait_asynccnt<5>();                           // tile t landed, t+1 in flight
    } else {
      wait_asynccnt<0>();
    }
    __syncthreads();                                // all waves' copies visible

    const float* buf = ldsbuf[t & 1];
#pragma unroll
    for (int step = 0; step < 2; ++step) {
      const int nl = wave * 4 + step * 2 + half;    // 0..31 within tile
      const float* vrow = buf + nl * CD;            // + c*16 + dd : conflict-free
      const float a = act_in[(size_t)b * NN + n0 + t * TILE_N + nl];

      float varr[CC];
      float r[CC];
      if (ITER0) {
        const float rv = (a * (1.0f / CC)) / (a + EPSF);
#pragma unroll
        for (int c = 0; c < CC; ++c) {
          varr[c] = vrow[c * DD + dd];
          r[c] = rv;
        }
      } else {
        float lps[CC];
#pragma unroll
        for (int c = 0; c < CC; ++c) {
          float v  = vrow[c * DD + dd];
          varr[c]  = v;
          float df = v - p_miu[c * DD + dd];
          lps[c]   = df * df * p_i2s[c * DD + dd];
        }
        // sum over d (16 lanes of this half) via xor butterflies
#pragma unroll
        for (int m = 1; m < 16; m <<= 1) {
#pragma unroll
          for (int c = 0; c < CC; ++c) lps[c] += __shfl_xor(lps[c], m, 32);
        }
        float S[CC];
        float mx = -3.0e38f;
#pragma unroll
        for (int c = 0; c < CC; ++c) {
          S[c] = -p_hls[c] - lps[c];                // Σ_d log_p (unshifted)
          mx   = fmaxf(mx, S[c]);
        }
        float ap[CC];
        float sum = 0.f;
#pragma unroll
        for (int c = 0; c < CC; ++c) {
          ap[c] = __expf(S[c] - mx) * p_act[c];
          sum  += ap[c];
        }
        const float inv = 1.f / (sum + EPSF);
        float sum2 = 0.f;
#pragma unroll
        for (int c = 0; c < CC; ++c) {
          r[c]  = ap[c] * inv * a;                  // r = r*activation
          sum2 += r[c];
        }
        const float inv2 = 1.f / (sum2 + EPSF);     // renormalize over C
#pragma unroll
        for (int c = 0; c < CC; ++c) r[c] *= inv2;
      }

#pragma unroll
      for (int c = 0; c < CC; ++c) {
        rs[c] += r[c];
        const float rv = r[c] * varr[c];
        s1[c] += rv;
        s2[c] += rv * varr[c];
      }
    }
    __syncthreads();                                // buffer reusable
  }

  // combine the two 16-lane halves (deterministic)
#pragma unroll
  for (int c = 0; c < CC; ++c) {
    s1[c] += __shfl_xor(s1[c], 16, 32);
    s2[c] += __shfl_xor(s2[c], 16, 32);
    rs[c] += __shfl_xor(rs[c], 16, 32);
  }
  if (lane < 16) {
#pragma unroll
    for (int c = 0; c < CC; ++c) {
      wred[wave][c * DD + dd]       = s1[c];
      wred[wave][160 + c * DD + dd] = s2[c];
    }
  }
  if (lane == 0) {
#pragma unroll
    for (int c = 0; c < CC; ++c) wred[wave][320 + c] = rs[c];
  }
  __syncthreads();
  // fixed-order cross-wave sum -> per-block partials (deterministic, no atomics)
  for (int i = tid; i < 330; i += 256) {
    float acc = 0.f;
#pragma unroll
    for (int w = 0; w < 8; ++w) acc += wred[w][i];
    partials[(size_t)blk * PSTRIDE + i] = acc;
  }
}

// ---------------- tiny epilogue: close the reductions, build next-iter params ----------------
__global__ __launch_bounds__(256)
void em_epilogue(const float* __restrict__ partials,
                 float* __restrict__ miu_p,
                 float* __restrict__ i2s_p,
                 float* __restrict__ hls_p,
                 float* __restrict__ act_p,
                 float* __restrict__ out,
                 int last) {
  const int b   = blockIdx.x;
  const int tid = threadIdx.x;
  __shared__ float rS1[CD], rS2[CD], rRS[CC], sig[CD];

  for (int i = tid; i < 330; i += 256) {
    float acc = 0.f;
    const float* p = partials + (size_t)b * CHUNKS * PSTRIDE + i;
    for (int k = 0; k < CHUNKS; ++k) acc += p[(size_t)k * PSTRIDE];  // fixed order
    if (i < 160)      rS1[i]       = acc;
    else if (i < 320) rS2[i - 160] = acc;
    else              rRS[i - 320] = acc;
  }
  __syncthreads();

  if (tid < CD) {
    const int   c    = tid >> 4;
    const float rsum = rRS[c];
    const float tden = rsum + EPSF;
    const float S1 = rS1[tid], S2 = rS2[tid];
    const float mu = S1 / tden;
    float sg = (S2 - mu * (2.f * S1 - mu * rsum)) / tden;  // Σ r1 (v-μ)^2
    sg = fmaxf(sg, 0.f) + EPSF;
    sig[tid] = sg;
    miu_p[b * CD + tid] = mu;
    i2s_p[b * CD + tid] = 0.5f / sg;
    if (last) out[b * CD + tid] = mu;                      // [B,1,C,D] flat
  }
  __syncthreads();

  if (tid < CC) {
    float h = 0.f;
#pragma unroll
    for (int d = 0; d < DD; ++d) h += 0.5f * __logf(sig[tid * DD + d]);
    hls_p[b * CC + tid] = h;
  }
  if (tid == 0) {
    float mx = -3.0e38f;
#pragma unroll
    for (int c = 0; c < CC; ++c) mx = fmaxf(mx, rRS[c]);
    float e[CC];
    float s = 0.f;
#pragma unroll
    for (int c = 0; c < CC; ++c) { e[c] = __expf(rRS[c] - mx); s += e[c]; }
    const float inv = 1.f / s;
#pragma unroll
    for (int c = 0; c < CC; ++c) act_p[b * CC + c] = e[c] * inv;  // softmax over C
  }
}

// ---------------- launcher ----------------
extern "C" void kernel_launch(void* const* d_in, const int* in_sizes, int n_in,
                              void* d_out, int out_size, void* d_ws, size_t ws_size,
                              hipStream_t stream) {
  const float* votes = (const float*)d_in[0];
  const float* act   = (const float*)d_in[1];
  // d_in[2] (beta_v) / d_in[3] (beta_a) only affect the final act_out, not miu.

  float* ws       = (float*)d_ws;
  float* partials = ws;                                  // 16*64*336 = 344064
  float* miu_p    = partials + (size_t)BB * CHUNKS * PSTRIDE;
  float* i2s_p    = miu_p + BB * CD;
  float* hls_p    = i2s_p + BB * CD;
  float* act_p    = hls_p + BB * CC;
  float* out      = (float*)d_out;

  const dim3 gp(BB * CHUNKS), bp(256);
  const dim3 ge(BB), be(256);

  // iter 0
  em_pass<true><<<gp, bp, 0, stream>>>(votes, act, miu_p, i2s_p, hls_p, act_p, partials);
  em_epilogue<<<ge, be, 0, stream>>>(partials, miu_p, i2s_p, hls_p, act_p, out, 0);
  // iter 1
  em_pass<false><<<gp, bp, 0, stream>>>(votes, act, miu_p, i2s_p, hls_p, act_p, partials);
  em_epilogue<<<ge, be, 0, stream>>>(partials, miu_p, i2s_p, hls_p, act_p, out, 0);
  // iter 2 (final): epilogue writes miu to d_out
  em_pass<false><<<gp, bp, 0, stream>>>(votes, act, miu_p, i2s_p, hls_p, act_p, partials);
  em_epilogue<<<ge, be, 0, stream>>>(partials, miu_p, i2s_p, hls_p, act_p, out, 1);
}